// Decoder_40570261078500
// MI455X (gfx1250) — compile-verified
//
#include <hip/hip_runtime.h>
#include <hip/hip_bf16.h>

// Problem constants (match reference)
#define BB 2
#define SS 1024
#define DD 512
#define CF 32
#define LL 4
#define DFF 2048
#define DI 1024
#define NN 16
#define KK 4
#define RR 32
#define MM (BB * SS)   // 2048 tokens

typedef float v2f __attribute__((ext_vector_type(2)));
typedef float v8f __attribute__((ext_vector_type(8)));

// ---------------------------------------------------------------------------
// Transpose uni_w (D, CF, D) [o][i][j] -> Wt2 interleaved [i][j/2][o][j%2]
// so a B fragment (two consecutive K values for one o) is one 8-byte load.
// ---------------------------------------------------------------------------
__global__ void k_transpose_uni(const float* __restrict__ uw, float* __restrict__ wt2) {
    int idx = blockIdx.x * 256 + threadIdx.x;   // over CF*D*D
    if (idx >= CF * DD * DD) return;
    int o = idx % DD;
    int j = (idx / DD) % DD;
    int i = idx / (DD * DD);
    size_t pos = (((size_t)i * (DD / 2) + (j >> 1)) * DD + o) * 2 + (j & 1);
    wt2[pos] = uw[((size_t)o * CF + i) * DD + j];
}

// ---------------------------------------------------------------------------
// h[m,o] = sum_i ctx[m,i] * (x[m,:] . uni_w[o,i,:]) + uni_b[o]
// One wave32 per 16(m) x 32(o) tile: two WMMA accumulators share each
// context-scaled A fragment; B fragments are single b64 loads.
// ---------------------------------------------------------------------------
__global__ void k_uni(const float* __restrict__ x, const float* __restrict__ ctx,
                      const float* __restrict__ wt2, const float* __restrict__ ub,
                      float* __restrict__ h) {
    __shared__ float Xs[16][516];   // stride 516 -> conflict-free column reads
    __shared__ float Cs[16][32];
    int t  = threadIdx.x;
    int o0 = blockIdx.x * 32;
    int m0 = blockIdx.y * 16;

    // Stage X tile (16 x 512 f32) and context tile (16 x 32) in LDS.
    for (int q = 0; q < 64; ++q) {
        int lin = q * 32 + t;          // float4 index, 2048 total
        int row = lin >> 7;            // 128 float4 per row
        int c4  = lin & 127;
        *reinterpret_cast<float4*>(&Xs[row][c4 * 4]) =
            *reinterpret_cast<const float4*>(&x[(size_t)(m0 + row) * DD + c4 * 4]);
    }
    for (int q = 0; q < 4; ++q) {
        int lin = q * 32 + t;          // 128 float4
        int row = lin >> 3;
        int c4  = lin & 7;
        *reinterpret_cast<float4*>(&Cs[row][c4 * 4]) =
            *reinterpret_cast<const float4*>(&ctx[(size_t)(m0 + row) * CF + c4 * 4]);
    }
    __syncthreads();

    int ml = t & 15;          // A row (m) / B col (o) within tile
    int kh = t >> 4;          // K-half selector per f32 WMMA layout
    v8f acc0 = {};
    v8f acc1 = {};
    for (int i = 0; i < CF; ++i) {
        float scale = Cs[ml][i];
        // interleaved layout: element (i, jj, o, p) at (((i*256 + jj/2)*512)+o)*2 + p
        const float* wpi = wt2 + ((size_t)i * 256 * 512 + (o0 + ml)) * 2;
        for (int j = 0; j < DD; j += 4) {
            int jj2 = (j >> 1) + kh;               // (j + 2*kh) / 2
            v2f a, b0, b1;
            a.x = Xs[ml][j + 2 * kh]     * scale;
            a.y = Xs[ml][j + 2 * kh + 1] * scale;
            b0 = *reinterpret_cast<const v2f*>(wpi + (size_t)jj2 * 1024);
            b1 = *reinterpret_cast<const v2f*>(wpi + (size_t)jj2 * 1024 + 32);
            acc0 = __builtin_amdgcn_wmma_f32_16x16x4_f32(
                false, a, false, b0, (short)0, acc0, false, false);
            acc1 = __builtin_amdgcn_wmma_f32_16x16x4_f32(
                false, a, false, b1, (short)0, acc1, false, false);
        }
    }
    int   n0i = o0 + ml;
    float bn0 = ub[n0i];
    float bn1 = ub[n0i + 16];
    for (int v = 0; v < 8; ++v) {
        int m = m0 + v + 8 * kh;
        h[(size_t)m * DD + n0i]      = acc0[v] + bn0;
        h[(size_t)m * DD + n0i + 16] = acc1[v] + bn1;
    }
}

// ---------------------------------------------------------------------------
// Generic WMMA GEMM: C[m,n] = act( sum_k A[m,k]*Bw[n,k] + bias[n] ) (+= C)
// One wave32 per 16(m) x 64(n) tile: 4 accumulators share each A fragment.
// K staged in LDS in chunks of 32.  act: 0=none, 1=relu, 2=softplus
// ---------------------------------------------------------------------------
__global__ void k_gemm(const float* __restrict__ A, int lda,
                       const float* __restrict__ Bw, int ldb,
                       float* __restrict__ C, int ldc,
                       const float* __restrict__ bias,
                       int Kd, int act, int accum) {
    __shared__ float As[16][36];
    __shared__ float Bs[64][36];
    int t  = threadIdx.x;
    int n0 = blockIdx.x * 64;
    int m0 = blockIdx.y * 16;
    int ml = t & 15;
    int kh = t >> 4;

    v8f acc[4] = {{}, {}, {}, {}};
    for (int k0 = 0; k0 < Kd; k0 += 32) {
        __syncthreads();
        for (int q = 0; q < 4; ++q) {            // A tile: 128 float4
            int lin = q * 32 + t;
            int row = lin >> 3;
            int c4  = lin & 7;
            *reinterpret_cast<float4*>(&As[row][c4 * 4]) =
                *reinterpret_cast<const float4*>(&A[(size_t)(m0 + row) * lda + k0 + c4 * 4]);
        }
        for (int q = 0; q < 16; ++q) {           // B tile: 512 float4
            int lin = q * 32 + t;
            int row = lin >> 3;
            int c4  = lin & 7;
            *reinterpret_cast<float4*>(&Bs[row][c4 * 4]) =
                *reinterpret_cast<const float4*>(&Bw[(size_t)(n0 + row) * ldb + k0 + c4 * 4]);
        }
        __syncthreads();
        for (int kk = 0; kk < 32; kk += 4) {
            int c = kk + 2 * kh;
            v2f a;
            a.x = As[ml][c];
            a.y = As[ml][c + 1];
#pragma unroll
            for (int nt = 0; nt < 4; ++nt) {
                v2f b;
                b.x = Bs[nt * 16 + ml][c];
                b.y = Bs[nt * 16 + ml][c + 1];
                acc[nt] = __builtin_amdgcn_wmma_f32_16x16x4_f32(
                    false, a, false, b, (short)0, acc[nt], false, false);
            }
        }
    }
#pragma unroll
    for (int nt = 0; nt < 4; ++nt) {
        int   n  = n0 + nt * 16 + ml;
        float bv = bias ? bias[n] : 0.0f;
        for (int v = 0; v < 8; ++v) {
            int   m   = m0 + v + 8 * kh;
            float val = acc[nt][v] + bv;
            if (act == 1)      val = fmaxf(val, 0.0f);
            else if (act == 2) val = (val > 20.0f) ? val : log1pf(__expf(val));
            if (accum) val += C[(size_t)m * ldc + n];
            C[(size_t)m * ldc + n] = val;
        }
    }
}

// ---------------------------------------------------------------------------
// LayerNorm over last dim (D=512). One block of 256 threads per row.
// ---------------------------------------------------------------------------
__global__ void k_ln(const float* __restrict__ h, const float* __restrict__ g,
                     const float* __restrict__ be, float* __restrict__ out) {
    __shared__ float sbuf[256];
    __shared__ float qbuf[256];
    int m = blockIdx.x;
    int t = threadIdx.x;
    float x0 = h[(size_t)m * DD + t];
    float x1 = h[(size_t)m * DD + t + 256];
    sbuf[t] = x0 + x1;
    qbuf[t] = x0 * x0 + x1 * x1;
    __syncthreads();
    for (int o = 128; o > 0; o >>= 1) {
        if (t < o) { sbuf[t] += sbuf[t + o]; qbuf[t] += qbuf[t + o]; }
        __syncthreads();
    }
    float mu  = sbuf[0] * (1.0f / DD);
    float var = qbuf[0] * (1.0f / DD) - mu * mu;
    float r   = rsqrtf(var + 1e-5f);
    out[(size_t)m * DD + t]       = (x0 - mu) * r * g[t]       + be[t];
    out[(size_t)m * DD + t + 256] = (x1 - mu) * r * g[t + 256] + be[t + 256];
}

// ---------------------------------------------------------------------------
// Depthwise causal conv (K=4) over xi = xz[..., :DI], then SiLU -> xa.
// ---------------------------------------------------------------------------
__global__ void k_conv(const float* __restrict__ xz, const float* __restrict__ cw,
                       const float* __restrict__ cb, float* __restrict__ xa) {
    int idx = blockIdx.x * 256 + threadIdx.x;
    if (idx >= MM * DI) return;
    int d = idx % DI;
    int m = idx / DI;
    int s = m % SS;
    const float* xi = xz + (size_t)m * (2 * DI) + d;
    float v = cb[d];
    if (s >= 3) v += cw[d * KK + 0] * xi[-3 * (2 * DI)];
    if (s >= 2) v += cw[d * KK + 1] * xi[-2 * (2 * DI)];
    if (s >= 1) v += cw[d * KK + 2] * xi[-1 * (2 * DI)];
    v += cw[d * KK + 3] * xi[0];
    xa[idx] = v * (1.0f / (1.0f + __expf(-v)));   // SiLU
}

// ---------------------------------------------------------------------------
// Selective scan: one thread per (b, di) channel, 16 states in registers.
// y = (scan_out + D*xa) * silu(z)
// ---------------------------------------------------------------------------
__global__ void k_scan(const float* __restrict__ dt, const float* __restrict__ xa,
                       const float* __restrict__ dbl, const float* __restrict__ xz,
                       const float* __restrict__ a_log, const float* __restrict__ dp,
                       float* __restrict__ y) {
    int di = blockIdx.x * 256 + threadIdx.x;
    int b  = blockIdx.y;
    float Ac[NN], st[NN];
    for (int n = 0; n < NN; ++n) {
        Ac[n] = -__expf(a_log[(size_t)di * NN + n]);
        st[n] = 0.0f;
    }
    float Dv = dp[di];
    for (int s = 0; s < SS; ++s) {
        size_t m   = (size_t)b * SS + s;
        float  dtv = dt[m * DI + di];
        float  xav = xa[m * DI + di];
        float  zv  = xz[m * (2 * DI) + DI + di];
        float  dx  = dtv * xav;
        const float* row = dbl + m * 64;
        float accv = 0.0f;
        for (int n = 0; n < NN; ++n) {
            float dA = __expf(dtv * Ac[n]);
            st[n] = st[n] * dA + dx * row[RR + n];        // B part
            accv += st[n] * row[RR + NN + n];             // C part
        }
        float sz = zv * (1.0f / (1.0f + __expf(-zv)));
        y[m * DI + di] = (accv + Dv * xav) * sz;
    }
}

// ---------------------------------------------------------------------------
// Host orchestration
// ---------------------------------------------------------------------------
extern "C" void kernel_launch(void* const* d_in, const int* in_sizes, int n_in,
                              void* d_out, int out_size, void* d_ws, size_t ws_size,
                              hipStream_t stream) {
    (void)in_sizes; (void)n_in; (void)out_size; (void)ws_size;

    const float* x        = (const float*)d_in[0];
    const float* context  = (const float*)d_in[1];
    const float* uni_w    = (const float*)d_in[2];
    const float* uni_b    = (const float*)d_in[3];
    const float* ln_g     = (const float*)d_in[4];
    const float* ln_b     = (const float*)d_in[5];
    const float* in_pw    = (const float*)d_in[6];
    const float* conv_w   = (const float*)d_in[7];
    const float* conv_b   = (const float*)d_in[8];
    const float* x_pw     = (const float*)d_in[9];
    const float* dt_pw    = (const float*)d_in[10];
    const float* dt_pb    = (const float*)d_in[11];
    const float* A_log    = (const float*)d_in[12];
    const float* D_param  = (const float*)d_in[13];
    const float* out_pw   = (const float*)d_in[14];
    const float* ff_w1    = (const float*)d_in[15];
    const float* ff_b1    = (const float*)d_in[16];
    const float* ff_w2    = (const float*)d_in[17];
    const float* ff_b2    = (const float*)d_in[18];

    float* h = (float*)d_out;   // (M, D) residual stream lives in d_out

    // Workspace carve-up (floats)
    float* ws  = (float*)d_ws;
    float* wt  = ws;                       size_t off = (size_t)CF * DD * DD;   // 8.4M
    float* hn  = ws + off;                 off += (size_t)MM * DD;              // 1.0M
    float* xz  = ws + off;                 off += (size_t)MM * 2 * DI;          // 4.2M
    float* xa  = ws + off;                 off += (size_t)MM * DI;              // 2.1M
    float* dbl = ws + off;                 off += (size_t)MM * 64;              // 0.13M
    float* dtb = ws + off;                 off += (size_t)MM * DI;              // 2.1M
    float* yb  = ws + off;                 off += (size_t)MM * DI;              // 2.1M
    float* ff1 = ws + off;                 off += (size_t)MM * DFF;             // 4.2M

    // --- uni projection: h = einsum pair + bias -----------------------------
    {
        int total = CF * DD * DD;
        k_transpose_uni<<<(total + 255) / 256, 256, 0, stream>>>(uni_w, wt);
        dim3 grid(DD / 32, MM / 16);
        k_uni<<<grid, 32, 0, stream>>>(x, context, wt, uni_b, h);
    }

    for (int l = 0; l < LL; ++l) {
        const float* g   = ln_g   + (size_t)l * DD;
        const float* be  = ln_b   + (size_t)l * DD;
        const float* inw = in_pw  + (size_t)l * 2 * DI * DD;
        const float* cw  = conv_w + (size_t)l * DI * KK;
        const float* cb  = conv_b + (size_t)l * DI;
        const float* xpw = x_pw   + (size_t)l * (RR + 2 * NN) * DI;
        const float* dtw = dt_pw  + (size_t)l * DI * RR;
        const float* dtp = dt_pb  + (size_t)l * DI;
        const float* al  = A_log  + (size_t)l * DI * NN;
        const float* dpr = D_param+ (size_t)l * DI;
        const float* ow  = out_pw + (size_t)l * DD * DI;
        const float* w1  = ff_w1  + (size_t)l * DFF * DD;
        const float* b1  = ff_b1  + (size_t)l * DFF;
        const float* w2  = ff_w2  + (size_t)l * DD * DFF;
        const float* b2  = ff_b2  + (size_t)l * DD;

        // hn = LN(h)
        k_ln<<<MM, 256, 0, stream>>>(h, g, be, hn);
        // xz = hn @ in_proj_w^T   (M x 2DI)
        k_gemm<<<dim3(2 * DI / 64, MM / 16), 32, 0, stream>>>(
            hn, DD, inw, DD, xz, 2 * DI, nullptr, DD, 0, 0);
        // xa = silu(conv(xz[:, :DI]))
        k_conv<<<(MM * DI + 255) / 256, 256, 0, stream>>>(xz, cw, cb, xa);
        // dbl = xa @ x_proj_w^T   (M x 64)
        k_gemm<<<dim3(64 / 64, MM / 16), 32, 0, stream>>>(
            xa, DI, xpw, DI, dbl, 64, nullptr, DI, 0, 0);
        // dt = softplus(dbl[:, :R] @ dt_proj_w^T + dt_proj_b)  (M x DI)
        k_gemm<<<dim3(DI / 64, MM / 16), 32, 0, stream>>>(
            dbl, 64, dtw, RR, dtb, DI, dtp, RR, 2, 0);
        // selective scan -> y
        k_scan<<<dim3(DI / 256, BB), 256, 0, stream>>>(
            dtb, xa, dbl, xz, al, dpr, yb);
        // h += y @ out_proj_w^T
        k_gemm<<<dim3(DD / 64, MM / 16), 32, 0, stream>>>(
            yb, DI, ow, DI, h, DD, nullptr, DI, 0, 1);
        // hn = LN(h)
        k_ln<<<MM, 256, 0, stream>>>(h, g, be, hn);
        // ff1 = relu(hn @ ff_w1^T + ff_b1)
        k_gemm<<<dim3(DFF / 64, MM / 16), 32, 0, stream>>>(
            hn, DD, w1, DD, ff1, DFF, b1, DD, 1, 0);
        // h += ff1 @ ff_w2^T + ff_b2
        k_gemm<<<dim3(DD / 64, MM / 16), 32, 0, stream>>>(
            ff1, DFF, w2, DFF, h, DD, b2, DFF, 0, 1);
    }
}